// SpikingKWTA_28552942584256
// MI455X (gfx1250) — compile-verified
//
#include <hip/hip_runtime.h>
#include <stdint.h>

// SpikingConfig constants (reference)
#define KW        5
#define GAIN_UP   1.5f
#define GAIN_DOWN 0.6f

typedef unsigned long long u64;

// ---------------------------------------------------------------------------
// Kernel 1: zero the count buffer (d_out reused as uint32 counts).
// ---------------------------------------------------------------------------
__global__ __launch_bounds__(256) void spk_zero_kernel(uint32_t* __restrict__ counts, int V) {
    int i  = blockIdx.x * blockDim.x + threadIdx.x;
    int i4 = i * 4;
    if (i4 + 3 < V) {
        uint4 z = {0u, 0u, 0u, 0u};
        *reinterpret_cast<uint4*>(counts + i4) = z;
    } else {
        for (int j = i4; j < V; ++j) counts[j] = 0u;
    }
}

// ---------------------------------------------------------------------------
// Kernel 2: histogram of token ids (spike count == occurrence count: with zero
// initial potentials every occurrence crosses threshold exactly once and the
// soft-reset residual is identically zero).
// Ids are staged into LDS with the gfx1250 async-copy path
// (global_load_async_to_lds_b128 + s_wait_asynccnt), then scattered with
// global atomics. Tail elements (L not a multiple of the tile) bypass LDS.
// ---------------------------------------------------------------------------
#define HIST_TILE 1024   // ids per block = 256 threads * 4 ids (16B) each

__global__ __launch_bounds__(256) void spk_hist_kernel(const int* __restrict__ ids,
                                                       uint32_t* __restrict__ counts,
                                                       int L, int V) {
    __shared__ __attribute__((aligned(16))) int s_ids[HIST_TILE];

    const int tid  = threadIdx.x;
    const int base = blockIdx.x * HIST_TILE + tid * 4;
    const bool full = (base + 3 < L);

    if (full) {
        // Flat shared pointers carry the wave-relative LDS byte address in
        // their low 32 bits (LDS aperture mapping), which is exactly what the
        // async instruction's LDS-address VGPR expects.
        uint32_t lds_off = (uint32_t)(uintptr_t)(void*)&s_ids[tid * 4];
        uint64_t gaddr   = (uint64_t)(uintptr_t)(ids + base);
        asm volatile("global_load_async_to_lds_b128 %0, %1, off"
                     :: "v"(lds_off), "v"(gaddr) : "memory");
    }
    // Wait for this wave's async transfers, then barrier for the other waves.
    asm volatile("s_wait_asynccnt 0x0" ::: "memory");
    __syncthreads();

    if (full) {
        #pragma unroll
        for (int j = 0; j < 4; ++j) {
            int t = s_ids[tid * 4 + j];
            if ((unsigned)t < (unsigned)V) atomicAdd(&counts[t], 1u);
        }
    } else {
        for (int j = 0; j < 4; ++j) {
            int i = base + j;
            if (i < L) {
                int t = ids[i];
                if ((unsigned)t < (unsigned)V) atomicAdd(&counts[t], 1u);
            }
        }
    }
}

// ---------------------------------------------------------------------------
// Kernel 3: top-5 by (count desc, index asc), single workgroup, SINGLE sweep.
// Key = (count << 32) | (0xFFFFFFFF - index): max key == jax.lax.top_k winner
// (stable lower-index tie-break). Each thread keeps a private sorted top-5 in
// registers (unrolled bubble insert), then a 5-round tournament: wave32
// shuffle max + one LDS ds_max_u64 per wave; the unique owner of the winning
// key (keys embed the index) pops its list head. Writes sel[0..4] and
// sel[5] = anyActive (max count > 0) straight to ws.
// ---------------------------------------------------------------------------
__global__ __launch_bounds__(1024) void spk_topk_kernel(const uint32_t* __restrict__ counts,
                                                        int V, int* __restrict__ sel_out) {
    __shared__ u64 blockMax;

    const int tid = threadIdx.x;
    const int nth = blockDim.x;
    const int k   = (KW < V) ? KW : V;

    u64 top[KW];
    #pragma unroll
    for (int j = 0; j < KW; ++j) top[j] = 0ull;

    for (int v = tid; v < V; v += nth) {
        __builtin_prefetch(counts + v + 4 * nth, 0, 0);   // global_prefetch_b8
        uint32_t c = counts[v];
        u64 key = ((u64)c << 32) | (u64)(0xFFFFFFFFu - (uint32_t)v);
        if (key > top[KW - 1]) {
            top[KW - 1] = key;
            #pragma unroll
            for (int j = KW - 1; j > 0; --j) {   // bubble the new key up
                u64 a = top[j - 1], b = top[j];
                top[j - 1] = (b > a) ? b : a;
                top[j]     = (b > a) ? a : b;
            }
        }
    }

    for (int pass = 0; pass < k; ++pass) {
        if (tid == 0) blockMax = 0ull;
        __syncthreads();

        const u64 cand = top[0];
        u64 red = cand;
        #pragma unroll
        for (int off = 16; off > 0; off >>= 1) {          // wave32 max-reduce
            u64 o = __shfl_xor(red, off, 32);
            red = (o > red) ? o : red;
        }
        if ((tid & 31) == 0) atomicMax(&blockMax, red);   // ds_max_u64
        __syncthreads();

        const u64 m = blockMax;
        if (cand == m && m != 0ull) {                     // unique owner pops
            #pragma unroll
            for (int j = 0; j < KW - 1; ++j) top[j] = top[j + 1];
            top[KW - 1] = 0ull;
        }
        if (tid == 0) {
            sel_out[pass] = (int)(0xFFFFFFFFu - (uint32_t)(m & 0xFFFFFFFFull));
            if (pass == 0) sel_out[KW] = ((m >> 32) != 0ull) ? 1 : 0;
        }
        __syncthreads();   // everyone done reading blockMax before next reset
    }

    if (tid == 0) {
        for (int pass = k; pass < KW; ++pass) sel_out[pass] = -1;
    }
}

// ---------------------------------------------------------------------------
// Kernel 4: write gains in place over the count buffer (d_out).
// gain = 1.0 if nothing spiked; else 1.5 for top-5, 0.6 for active, 1.0 idle.
// ---------------------------------------------------------------------------
__global__ __launch_bounds__(256) void spk_gains_kernel(uint32_t* __restrict__ counts_inout,
                                                        const int* __restrict__ sel, int V) {
    int v = blockIdx.x * blockDim.x + threadIdx.x;
    if (v >= V) return;

    uint32_t c = counts_inout[v];
    // uniform scalar loads (broadcast)
    int s0 = sel[0], s1 = sel[1], s2 = sel[2], s3 = sel[3], s4 = sel[4];
    int anyActive = sel[5];

    float g;
    if (!anyActive) {
        g = 1.0f;
    } else if (v == s0 || v == s1 || v == s2 || v == s3 || v == s4) {
        g = GAIN_UP;
    } else {
        g = (c > 0u) ? GAIN_DOWN : 1.0f;
    }
    reinterpret_cast<float*>(counts_inout)[v] = g;
}

// ---------------------------------------------------------------------------
// Launch: d_in[0]=token_ids (L ints), d_in[1]=vocab_size (scalar),
// d_in[2]=potentials_init (V floats, zeros). d_out (V floats) doubles as the
// uint32 count buffer before being overwritten with the gains.
// ---------------------------------------------------------------------------
extern "C" void kernel_launch(void* const* d_in, const int* in_sizes, int n_in,
                              void* d_out, int out_size, void* d_ws, size_t ws_size,
                              hipStream_t stream) {
    const int* ids = (const int*)d_in[0];
    const int L = in_sizes[0];
    const int V = in_sizes[2];   // length of potentials_init == vocab size

    uint32_t* counts = (uint32_t*)d_out;   // reuse output buffer as counts
    int*      sel    = (int*)d_ws;         // 6 ints: 5 winners + anyActive flag

    int zthreads = (V + 3) / 4;
    spk_zero_kernel<<<(zthreads + 255) / 256, 256, 0, stream>>>(counts, V);
    spk_hist_kernel<<<(L + HIST_TILE - 1) / HIST_TILE, 256, 0, stream>>>(ids, counts, L, V);
    spk_topk_kernel<<<1, 1024, 0, stream>>>(counts, V, sel);
    spk_gains_kernel<<<(V + 255) / 256, 256, 0, stream>>>(counts, sel, V);
}